// WRA_16037407883795
// MI455X (gfx1250) — compile-verified
//
#include <hip/hip_runtime.h>
#include <math.h>

typedef float v2f __attribute__((ext_vector_type(2)));
typedef float v8f __attribute__((ext_vector_type(8)));
typedef int   b128v __attribute__((vector_size(16)));   // matches builtin's V4i pointee

// ---- async-to-LDS availability (probe via __has_builtin; sync fallback) ----
#if defined(__has_builtin)
#if __has_builtin(__builtin_amdgcn_global_load_async_to_lds_b128) && \
    __has_builtin(__builtin_amdgcn_s_wait_asynccnt)
#define USE_ASYNC_LDS 1
#endif
#endif
#ifndef USE_ASYNC_LDS
#define USE_ASYNC_LDS 0
#endif

// Problem constants (match reference shapes)
constexpr int TL = 128, IL = 100, D = 768, NROWS = TL + IL;   // 228
constexpr int NT_N   = 7;          // ceil(100/16) n-tiles
constexpr int STRIDE = TL + 1;     // 129: A/T row stride -> conflict-free column walks
constexpr int K_CH   = 16;         // K-chunk per staging step
constexpr int NCH    = D / K_CH;   // 48 chunks
constexpr int STG    = 20;         // staging row stride (floats): 16B-aligned rows, no bank conflicts

// LDS layout (float offsets)
constexpr int OFF_A    = 0;                         // A: 100 x 129
constexpr int OFF_T    = OFF_A + IL * STRIDE;       // T: 100 x 129
constexpr int OFF_STG  = OFF_T + IL * STRIDE;       // double-buffered staging
constexpr int AT_SZ    = TL * STG;                  // 2560 floats
constexpr int BT_SZ    = IL * STG;                  // 2000 floats
constexpr int BUF_SZ   = AT_SZ + BT_SZ;             // 4560 floats per buffer
constexpr int OFF_INV  = OFF_STG + 2 * BUF_SZ;      // inv norms: 228 (reused as partials later)
constexpr int OFF_DEL  = OFF_INV + NROWS;           // delta: 100
constexpr int OFF_SIG  = OFF_DEL + IL;              // sigma: 128
constexpr int OFF_XM   = OFF_SIG + TL;              // x mask: 128
constexpr int OFF_YM   = OFF_XM + TL;               // y mask: 100
constexpr int OFF_RED  = OFF_YM + IL;               // reduce scratch: 8
constexpr int OFF_SCAL = OFF_RED + 8;               // xl, yl
constexpr int SMEM_FLOATS = OFF_SCAL + 2;
constexpr size_t SMEM_BYTES = (size_t)SMEM_FLOATS * sizeof(float);

__device__ __forceinline__ float waveReduce(float v) {
#pragma unroll
    for (int m = 16; m >= 1; m >>= 1) v += __shfl_xor(v, m, 32);
    return v;
}

#if USE_ASYNC_LDS
__device__ __forceinline__ void async_copy16(const float* gsrc, float* ldst) {
    __builtin_amdgcn_global_load_async_to_lds_b128(
        (__attribute__((address_space(1))) b128v*)gsrc,
        (__attribute__((address_space(3))) b128v*)ldst,
        0, 0);
}
#endif

// Stage one K-chunk (txt 128x16 + img 100x16, row-major stride 20) into buffer `buf`.
// Exactly 4 issue slots per thread -> uniform per-wave ASYNCcnt of 4 per chunk.
__device__ __forceinline__ void stage_chunk(const float* __restrict__ seqb,
                                            float* __restrict__ sm,
                                            int kc, int buf, int tid)
{
    float* At = sm + OFF_STG + buf * BUF_SZ;
    float* Bt = At + AT_SZ;
#pragma unroll
    for (int it = 0; it < 4; ++it) {
        int flat = tid + it * 256;            // 0..1023 slots; 512 txt + 400 img + clamp
        const float* gsrc;
        float* ldst;
        if (flat < 512) {                     // txt: 128 rows x 4 float4
            int m = flat >> 2, q = flat & 3;
            gsrc = seqb + (size_t)m * D + kc + q * 4;
            ldst = At + m * STG + q * 4;
        } else {                              // img: 100 rows x 4 float4 (tail clamped, benign dup)
            int u = flat - 512; if (u > 399) u = 399;
            int n = u >> 2, q = u & 3;
            gsrc = seqb + (size_t)(TL + n) * D + kc + q * 4;
            ldst = Bt + n * STG + q * 4;
        }
#if USE_ASYNC_LDS
        async_copy16(gsrc, ldst);
#else
        float a0 = gsrc[0], a1 = gsrc[1], a2 = gsrc[2], a3 = gsrc[3];
        ldst[0] = a0; ldst[1] = a1; ldst[2] = a2; ldst[3] = a3;
#endif
    }
}

__device__ __forceinline__ void wait_async_le4() {
#if USE_ASYNC_LDS
    __builtin_amdgcn_s_wait_asynccnt(4);
#endif
}
__device__ __forceinline__ void wait_async_le0() {
#if USE_ASYNC_LDS
    __builtin_amdgcn_s_wait_asynccnt(0);
#endif
}

__global__ __launch_bounds__(256, 1)
void ot_ipot_kernel(const float* __restrict__ seq,
                    const unsigned char* __restrict__ txt_pad,
                    const unsigned char* __restrict__ img_pad,
                    const int* __restrict__ is_correct,
                    float* __restrict__ ws)
{
    extern __shared__ float sm[];
    float* Ash  = sm + OFF_A;
    float* Tsh  = sm + OFF_T;
    float* invn = sm + OFF_INV;
    float* del  = sm + OFF_DEL;
    float* sig  = sm + OFF_SIG;
    float* xm   = sm + OFF_XM;
    float* ym   = sm + OFF_YM;
    float* red  = sm + OFF_RED;
    float* scal = sm + OFF_SCAL;

    const int b    = blockIdx.x;
    const int tid  = threadIdx.x;
    const int wv   = tid >> 5;     // wave id 0..7 == m-tile
    const int lane = tid & 31;
    const int hi   = lane >> 4;    // lane half (WMMA fragment mapping)
    const int nl   = lane & 15;

    const float* seqb = seq + (size_t)b * NROWS * D;

    // Kick off async staging of chunk 0 immediately (overlaps the norm pass).
    stage_chunk(seqb, sm, 0, 0, tid);

    // ---- masks & effective lengths ----
    if (tid < TL) xm[tid] = txt_pad[b * TL + tid] ? 10000.0f : 0.0f;
    if (tid < IL) ym[tid] = img_pad[b * IL + tid] ? 10000.0f : 0.0f;
    if (tid == 0) {
        int cx = 0, cy = 0;
        for (int m = 0; m < TL; ++m) cx += txt_pad[b * TL + m] ? 1 : 0;
        for (int n = 0; n < IL; ++n) cy += img_pad[b * IL + n] ? 1 : 0;
        scal[0] = (float)(TL - cx);
        scal[1] = (float)(IL - cy);
    }

    // ---- per-row inverse L2 norms (txt rows 0..127, img rows 128..227) ----
    for (int r = wv; r < NROWS; r += 8) {
        const float* row = seqb + (size_t)r * D;
        float s = 0.0f;
#pragma unroll 4
        for (int j = 0; j < D / 32; ++j) { float v = row[lane + 32 * j]; s += v * v; }
        s = waveReduce(s);
        if (lane == 0) invn[r] = 1.0f / fmaxf(sqrtf(s), 1e-5f);
    }
    __syncthreads();

    const float xl = scal[0], yl = scal[1];
    if (tid < TL) sig[tid] = (xm[tid] > 0.0f) ? 0.0f : (1.0f / xl);

    // ---- GEMM: G = txt(128xD) * img(100xD)^T via V_WMMA_F32_16X16X4_F32,
    //      double-buffered async LDS staging ----
    v8f acc[NT_N];
#pragma unroll
    for (int t = 0; t < NT_N; ++t) acc[t] = (v8f)0.0f;

    for (int c = 0; c < NCH; ++c) {
        if (c + 1 < NCH) {
            stage_chunk(seqb, sm, (c + 1) * K_CH, (c + 1) & 1, tid);
            wait_async_le4();            // older chunk (c) complete; chunk c+1 in flight
        } else {
            wait_async_le0();
        }
        __syncthreads();                 // chunk c visible to all waves

        const float* At = sm + OFF_STG + (c & 1) * BUF_SZ;
        const float* Bt = At + AT_SZ;

        // A fragments: lane<16 -> K=k0,k0+1 ; lane>=16 -> K=k0+2,k0+3 (same 16 rows)
        v2f afr[4];
        const float* ab = At + (wv * 16 + nl) * STG + hi * 2;
#pragma unroll
        for (int ks = 0; ks < 4; ++ks) afr[ks] = *(const v2f*)(ab + ks * 4);

#pragma unroll
        for (int nt = 0; nt < NT_N; ++nt) {
            const float* bb = Bt + (nt * 16 + nl) * STG + hi * 2;
#pragma unroll
            for (int ks = 0; ks < 4; ++ks) {
                v2f bfr = *(const v2f*)(bb + ks * 4);
                acc[nt] = __builtin_amdgcn_wmma_f32_16x16x4_f32(
                    false, afr[ks], false, bfr, (short)0, acc[nt], false, false);
            }
        }
        __syncthreads();                 // all reads of this buffer done before it is re-staged
    }

    // ---- epilogue: A = exp(-C/beta) = exp(2*ghat - 2), T0 = 1 (masked -> 0) ----
#pragma unroll
    for (int nt = 0; nt < NT_N; ++nt) {
        int n = nt * 16 + nl;
        if (n < IL) {
            float iny = invn[TL + n];
            bool  mn  = ym[n] > 0.0f;
#pragma unroll
            for (int i = 0; i < 8; ++i) {
                int   m      = wv * 16 + i + hi * 8;       // C/D fragment row mapping
                float ghat   = acc[nt][i] * invn[m] * iny; // cosine similarity
                bool  masked = mn || (xm[m] > 0.0f);
                int   off    = n * STRIDE + m;
                Ash[off] = masked ? 0.0f : expf(2.0f * ghat - 2.0f);
                Tsh[off] = masked ? 0.0f : 1.0f;
            }
        }
    }
    __syncthreads();

    // ---- IPOT: 50 Sinkhorn-style iterations, fully LDS-resident ----
    for (int it = 0; it < 50; ++it) {
        // qs[n] = sum_m A*T*sigma ; delta[n] = 1/(yl*qs + ymask)
        for (int n = wv; n < IL; n += 8) {
            const float* Ar = Ash + n * STRIDE;
            const float* Tr = Tsh + n * STRIDE;
            float s = 0.0f;
#pragma unroll
            for (int j = 0; j < 4; ++j) {
                int m = lane + 32 * j;
                s += Ar[m] * Tr[m] * sig[m];
            }
            s = waveReduce(s);
            if (lane == 0) del[n] = 1.0f / (yl * s + ym[n]);
        }
        __syncthreads();
        // sigma[m] = 1/(xl * sum_n delta*A*T + xmask) -- all 256 threads (n-range split)
        {
            int m    = tid & (TL - 1);
            int half = tid >> 7;
            float s = 0.0f;
            int n0 = half * 50;
            for (int n = n0; n < n0 + 50; ++n) {
                int off = n * STRIDE + m;
                s += del[n] * Ash[off] * Tsh[off];
            }
            if (half) invn[m] = s;        // invn reused as partial buffer (free after epilogue)
            __syncthreads();
            if (!half) sig[m] = 1.0f / (xl * (s + invn[m]) + xm[m]);
        }
        __syncthreads();
        // T = delta (.) (A*T) (.) sigma
        for (int idx = tid; idx < IL * TL; idx += 256) {
            int n = idx >> 7, m = idx & (TL - 1);
            int off = n * STRIDE + m;
            Tsh[off] = del[n] * Ash[off] * Tsh[off] * sig[m];
        }
        __syncthreads();
    }

    // ---- ot_dist[b] = sum_{m,n} cost * T ; cost = -beta*ln(A) (A==0 <=> masked) ----
    float part = 0.0f;
    for (int idx = tid; idx < IL * TL; idx += 256) {
        int n = idx >> 7, m = idx & (TL - 1);
        int off = n * STRIDE + m;
        float a = Ash[off];
        if (a > 0.0f) part += (-0.5f) * logf(a) * Tsh[off];
    }
    part = waveReduce(part);
    if (lane == 0) red[wv] = part;
    __syncthreads();
    if (tid == 0) {
        float s = 0.0f;
#pragma unroll
        for (int i = 0; i < 8; ++i) s += red[i];
        ws[b] = (is_correct[b] == 1) ? s : -s;   // signed contribution
    }
}

// Deterministic fixed-order final reduction: loss = sum(ws)/B
__global__ __launch_bounds__(256) void finalize_kernel(const float* __restrict__ ws,
                                                       float* __restrict__ out, int n)
{
    __shared__ float red[8];
    int tid = threadIdx.x;
    float v = (tid < n) ? ws[tid] : 0.0f;
    v = waveReduce(v);
    if ((tid & 31) == 0) red[tid >> 5] = v;
    __syncthreads();
    if (tid == 0) {
        float s = 0.0f;
#pragma unroll
        for (int i = 0; i < 8; ++i) s += red[i];
        out[0] = s / (float)n;
    }
}

extern "C" void kernel_launch(void* const* d_in, const int* in_sizes, int n_in,
                              void* d_out, int out_size, void* d_ws, size_t ws_size,
                              hipStream_t stream)
{
    const float*         seq        = (const float*)d_in[0];
    // d_in[1] = input_ids (unused), d_in[2] = image_feat (unused by reference)
    const unsigned char* txt_pad    = (const unsigned char*)d_in[3];
    const unsigned char* img_pad    = (const unsigned char*)d_in[4];
    const int*           is_correct = (const int*)d_in[5];
    float*               ws         = (float*)d_ws;
    float*               out        = (float*)d_out;

    const int B = in_sizes[5];   // 256

    ot_ipot_kernel<<<B, 256, SMEM_BYTES, stream>>>(seq, txt_pad, img_pad, is_correct, ws);
    finalize_kernel<<<1, 256, 0, stream>>>(ws, out, B);
}